// model_1_38783554683261
// MI455X (gfx1250) — compile-verified
//
#include <hip/hip_runtime.h>
#include <math.h>

// Problem constants (from reference)
#define B_  16
#define G_  4
#define D_  64
#define L_  4096
#define C_  128
#define E_  8
#define LP_ 4094

#define TL     64     // L-tile width
#define XSTR   66     // x tile LDS row stride (TL+2)
#define HSTR   65     // h tile LDS row stride (padded to dodge bank conflicts)
#define W1STR  192    // D*3 im2col K depth
#define TILES_PER_BLOCK 8

#define CF_SIZE   (B_*G_*C_*LP_)     // 33538048
#define LOSS_IDX  CF_SIZE
#define GATES_OFF (CF_SIZE + 1)

typedef __attribute__((ext_vector_type(2))) float v2f;
typedef __attribute__((ext_vector_type(8))) float v8f;
typedef __attribute__((ext_vector_type(4))) unsigned int u32x4;
typedef __attribute__((ext_vector_type(8))) int i32x8;
typedef __attribute__((ext_vector_type(4))) int i32x4;

#define XS_BYTE_OFF ((unsigned)((C_ * W1STR + C_) * 4))   // byte offset of xs in dynamic LDS

#if __has_builtin(__builtin_amdgcn_tensor_load_to_lds) && __has_builtin(__builtin_amdgcn_s_wait_tensorcnt)
#define USE_TDM 1
#else
#define USE_TDM 0
#endif

// Branchless tanh: 1 - 2/(exp(2x)+1), via v_exp_f32 (exp2) + v_rcp_f32.
// Saturates correctly: x>>0 -> exp2->inf -> rcp->0 -> +1 ; x<<0 -> exp2->0 -> rcp(1)=1 -> -1.
__device__ __forceinline__ float fast_tanh(float v) {
  float e = __builtin_amdgcn_exp2f(v * 2.8853900817779268f);  // 2*log2(e)
  return 1.0f - 2.0f * __builtin_amdgcn_rcpf(e + 1.0f);
}

#if USE_TDM
// Issue one 2D TDM tile load: 64 rows x 66 fp32 elements, global row stride 4096,
// into LDS at XS_BYTE_OFF (packed rows, stride 66). tensor_dim0 = elements remaining
// in the row from the tile start so the right-edge overrun reads back zeros (OOB rule).
__device__ __forceinline__ void tdm_load_xtile(const float* gsrc, unsigned rem_elems) {
  unsigned long long ga = (unsigned long long)(__SIZE_TYPE__)gsrc;
  u32x4 g0;
  g0[0] = 1u;                                   // count=1, user descriptor, no gather
  g0[1] = XS_BYTE_OFF;                          // lds_addr
  g0[2] = (unsigned)(ga & 0xffffffffu);         // global_addr[31:0]
  g0[3] = (unsigned)((ga >> 32) & 0x1ffffffu)   // global_addr[56:32]
        | 0x80000000u;                          // type=2 ("image")
  i32x8 g1;
  g1[0] = 0x00020000;                           // workgroup_mask=0, data_size=2 (4B)
  g1[1] = (int)((rem_elems & 0xffffu) << 16);   // tensor_dim0[15:0]
  g1[2] = (int)(((rem_elems >> 16) & 0xffffu)   // tensor_dim0[31:16]
        | (64u << 16));                         // tensor_dim1 = 64 rows
  g1[3] = (int)(66u << 16);                     // tensor_dim1 hi=0 | tile_dim0 = 66
  g1[4] = 64;                                   // tile_dim1 = 64, tile_dim2 = 0
  g1[5] = 4096;                                 // tensor_dim0_stride[31:0]
  g1[6] = 0;                                    // stride0 hi | stride1 lo
  g1[7] = 0;
  i32x4 gz4 = {0, 0, 0, 0};                     // groups 2/3 unused (2D tensor)
  i32x8 gz8 = {0, 0, 0, 0, 0, 0, 0, 0};
  __builtin_amdgcn_tensor_load_to_lds(g0, g1, gz4, gz4, gz8, 0);  // 6-arg toolchain form
}
#endif

// Workspace (floats): [0..255] topinfo[g][b][4] = {e0,e1,g0,g1}; [256..259] per-group loss
// ---------------------------------------------------------------------------
// Kernel A: gating (clean/noisy logits, softmax, top-3, gates, cdf load, loss)
// ---------------------------------------------------------------------------
__global__ void gating_kernel(const float* __restrict__ x,
                              const float* __restrict__ noise,
                              const float* __restrict__ wg,
                              const float* __restrict__ wn,
                              float* __restrict__ out,
                              float* __restrict__ ws) {
  const int g = blockIdx.x;
  __shared__ float cleanS[B_][E_], stdS[B_][E_], noisyS[B_][E_];
  __shared__ float probS[B_][E_], gateS[B_][E_];
  const int t = threadIdx.x;

  if (t < B_ * E_) {
    int b = t >> 3, e = t & 7;
    float cl = 0.f, rn = 0.f;
    const float* xb = x + ((size_t)(b * (G_ * D_) + g * D_)) * (size_t)L_ + (L_ - 6);
    for (int d = 0; d < D_; ++d) {
      for (int p = 0; p < 5; ++p) {
        float gv = xb[(size_t)d * L_ + p];
        int i = d * 5 + p;
        cl = fmaf(gv, wg[((size_t)g * (D_ * 5) + i) * E_ + e], cl);
        rn = fmaf(gv, wn[((size_t)g * (D_ * 5) + i) * E_ + e], rn);
      }
    }
    float sp = fmaxf(rn, 0.f) + log1pf(expf(-fabsf(rn)));  // stable softplus
    float sd = sp + 0.01f;
    float nz = noise[(g * B_ + b) * E_ + e];
    cleanS[b][e] = cl;
    stdS[b][e]   = sd;
    noisyS[b][e] = cl + nz * sd;
  }
  __syncthreads();

  if (t < B_) {
    const int b = t;
    float nv[E_], pv[E_];
    float m = -1e30f;
    for (int e = 0; e < E_; ++e) { nv[e] = noisyS[b][e]; m = fmaxf(m, nv[e]); }
    float s = 0.f;
    for (int e = 0; e < E_; ++e) { pv[e] = expf(nv[e] - m); s += pv[e]; }
    for (int e = 0; e < E_; ++e) pv[e] /= s;
    // top-3 of probs (ties -> lowest index, like jax top_k)
    int i0 = 0, i1 = -1; float v0 = -1.f, v1 = -1.f, v2 = -1.f;
    for (int e = 0; e < E_; ++e) if (pv[e] > v0) { v0 = pv[e]; i0 = e; }
    for (int e = 0; e < E_; ++e) if (e != i0 && pv[e] > v1) { v1 = pv[e]; i1 = e; }
    for (int e = 0; e < E_; ++e) if (e != i0 && e != i1 && pv[e] > v2) { v2 = pv[e]; }
    float denom = v0 + v1 + 1e-6f;
    float g0 = v0 / denom, g1 = v1 / denom;
    for (int e = 0; e < E_; ++e) gateS[b][e] = 0.f;
    gateS[b][i0] = g0; gateS[b][i1] = g1;
    const float th_in = v2, th_out = v1;     // prob-space thresholds (as in reference)
    const float INV_SQRT2 = 0.70710678118654752440f;
    for (int e = 0; e < E_; ++e) {
      float th = (nv[e] > th_in) ? th_in : th_out;
      float z = (cleanS[b][e] - th) / stdS[b][e];
      probS[b][e] = 0.5f * (1.f + erff(z * INV_SQRT2));
    }
    for (int e = 0; e < E_; ++e)
      out[GATES_OFF + (size_t)(b * E_ + e) * G_ + g] = gateS[b][e];
    float* ti = ws + (g * B_ + b) * 4;
    ti[0] = (float)i0; ti[1] = (float)i1; ti[2] = g0; ti[3] = g1;
  }
  __syncthreads();

  if (t == 0) {
    float imp[E_], ld[E_];
    for (int e = 0; e < E_; ++e) {
      float si = 0.f, sl = 0.f;
      for (int b = 0; b < B_; ++b) { si += gateS[b][e]; sl += probS[b][e]; }
      imp[e] = si; ld[e] = sl;
    }
    float loss = 0.f;
    for (int which = 0; which < 2; ++which) {
      const float* v = which ? ld : imp;
      float mean = 0.f;
      for (int e = 0; e < E_; ++e) mean += v[e];
      mean *= (1.f / E_);
      float var = 0.f;
      for (int e = 0; e < E_; ++e) { float d = v[e] - mean; var += d * d; }
      var *= (1.f / (E_ - 1));               // ddof=1
      loss += var / (mean * mean + 1e-10f);
    }
    ws[256 + g] = loss * 0.01f;
  }
}

// ---------------------------------------------------------------------------
// Kernel B: reduce per-group loss
// ---------------------------------------------------------------------------
__global__ void loss_sum_kernel(const float* __restrict__ ws, float* __restrict__ out) {
  out[LOSS_IDX] = ws[256] + ws[257] + ws[258] + ws[259];
}

// ---------------------------------------------------------------------------
// Kernel C: fused conv1 -> tanh -> (gate-folded conv2) via V_WMMA_F32_16X16X4_F32
// grid = 512 blocks (b,g,chunk), 256 threads = 8 waves. TDM streams x tiles,
// overlapped with stage-2 WMMA of the previous tile.
// ---------------------------------------------------------------------------
__global__ __launch_bounds__(256) void fused_conv_kernel(
    const float* __restrict__ x,
    const float* __restrict__ w1, const float* __restrict__ b1,
    const float* __restrict__ w2, const float* __restrict__ b2,
    const float* __restrict__ ws, float* __restrict__ out) {
  extern __shared__ float smem[];
  float* W1s = smem;                       // 128*192
  float* b1s = W1s + C_ * W1STR;           // 128
  float* xs  = b1s + C_;                   // 64*66   (byte offset XS_BYTE_OFF)
  float* hs  = xs + D_ * XSTR;             // 128*65  (h tile, then output staging)

  const int blk   = blockIdx.x;
  const int chunk = blk & 7;
  const int bg    = blk >> 3;
  const int b     = bg >> 2;
  const int g     = bg & 3;
  const int tid   = threadIdx.x;
  const int wave  = tid >> 5;
  const int lid   = tid & 31;
  const int half  = lid >> 4;    // 0: lanes 0-15, 1: lanes 16-31
  const int n     = lid & 15;    // M/N index within fragment
  const int kr0   = half * 2;    // K-row offset of this half-wave

  // stage conv1 weights (im2col-ordered [oc][d*3+t]) + bias into LDS
  for (int i = tid; i < C_ * W1STR; i += 256)
    W1s[i] = w1[(size_t)g * C_ * W1STR + i];
  for (int i = tid; i < C_; i += 256)
    b1s[i] = b1[g * C_ + i];

  const size_t xbase = ((size_t)(b * (G_ * D_) + g * D_)) * (size_t)L_;
  const size_t obase = ((size_t)(b * G_ + g)) * C_ * (size_t)LP_;
  const int l0base = chunk * TILES_PER_BLOCK * TL;

#if USE_TDM
  if (wave == 0)
    tdm_load_xtile(x + xbase + l0base, (unsigned)(L_ - l0base));
#endif

  // top-2 expert info for this (b,g)
  const float* ti = ws + (g * B_ + b) * 4;
  const int   e0 = (int)ti[0], e1 = (int)ti[1];
  const float g0 = ti[2],      g1 = ti[3];

  const int c0 = wave * 16;          // stage-1 M rows (conv1 out channels)
  const int d0 = wave * 16;          // stage-2 M rows (final d channels)
  const int dm = d0 + n;

  // Per-lane w_eff A-fragments in registers (top-2 gates folded into conv2 weights)
  v2f weff[32];
  const float* w2g = w2 + (size_t)g * (C_ * E_) * C_;
  #pragma unroll
  for (int kk = 0; kk < 32; ++kk) {
    int c = kk * 4 + kr0;
    float ax = g0 * w2g[(dm * E_ + e0) * C_ + c]     + g1 * w2g[(dm * E_ + e1) * C_ + c];
    float ay = g0 * w2g[(dm * E_ + e0) * C_ + c + 1] + g1 * w2g[(dm * E_ + e1) * C_ + c + 1];
    v2f w; w.x = ax; w.y = ay;
    weff[kk] = w;
  }
  const float beff = g0 * b2[g * (C_ * E_) + dm * E_ + e0]
                   + g1 * b2[g * (C_ * E_) + dm * E_ + e1];

  const v8f vzero = {0.f, 0.f, 0.f, 0.f, 0.f, 0.f, 0.f, 0.f};

  for (int tIdx = 0; tIdx < TILES_PER_BLOCK; ++tIdx) {
    const int l0 = l0base + tIdx * TL;

#if USE_TDM
    if (wave == 0)
      __builtin_amdgcn_s_wait_tensorcnt(0);   // xs(tile t) landed
#else
    for (int i = tid; i < D_ * XSTR; i += 256) {
      int r = i / XSTR, cc = i - r * XSTR;
      int l = l0 + cc;
      xs[i] = (l < L_) ? x[xbase + (size_t)r * L_ + l] : 0.f;
    }
    if (l0 + TL < L_) {
      const float* p = &x[xbase + (size_t)(tid & 63) * L_ + (l0 + TL) + ((tid >> 6) << 4)];
      __builtin_prefetch(p, 0, 1);
    }
#endif
    __syncthreads();  // B0: xs ready; prior iteration's staging reads done

    // --- stage 1: h = tanh(W1 (128x192) @ xcol (192x64) + b1)
    v8f acc[4];
    #pragma unroll
    for (int nt = 0; nt < 4; ++nt) acc[nt] = vzero;

    for (int kk = 0; kk < 48; ++kk) {
      const int ka = kk * 4 + kr0;
      const int kb = ka + 1;
      v2f a;
      a.x = W1s[(c0 + n) * W1STR + ka];
      a.y = W1s[(c0 + n) * W1STR + kb];
      const int da = ka / 3, ta = ka - da * 3;
      const int db = kb / 3, tb = kb - db * 3;
      #pragma unroll
      for (int nt = 0; nt < 4; ++nt) {
        const int col = nt * 16 + n;
        v2f bb;
        bb.x = xs[da * XSTR + col + ta];
        bb.y = xs[db * XSTR + col + tb];
        acc[nt] = __builtin_amdgcn_wmma_f32_16x16x4_f32(
            false, a, false, bb, (short)0, acc[nt], false, false);
      }
    }

    // bias + branchless tanh, write h tile to LDS
    #pragma unroll
    for (int nt = 0; nt < 4; ++nt) {
      #pragma unroll
      for (int i = 0; i < 8; ++i) {
        const int row = c0 + half * 8 + i;
        hs[row * HSTR + nt * 16 + n] = fast_tanh(acc[nt][i] + b1s[row]);
      }
    }
    __syncthreads();  // B1: hs ready; all xs reads of this tile done

#if USE_TDM
    // Overlap next tile's DMA with stage-2 compute
    if (wave == 0 && tIdx + 1 < TILES_PER_BLOCK) {
      const int ln = l0 + TL;
      tdm_load_xtile(x + xbase + ln, (unsigned)(L_ - ln));
    }
#endif

    // --- stage 2: moe = w_eff (128x128) @ h (128x64) + b_eff
    v8f acc2[4];
    #pragma unroll
    for (int nt = 0; nt < 4; ++nt) acc2[nt] = vzero;

    #pragma unroll
    for (int kk = 0; kk < 32; ++kk) {
      const int kb = kk * 4 + kr0;
      const v2f a = weff[kk];
      #pragma unroll
      for (int nt = 0; nt < 4; ++nt) {
        v2f bb;
        bb.x = hs[kb * HSTR + nt * 16 + n];
        bb.y = hs[(kb + 1) * HSTR + nt * 16 + n];
        acc2[nt] = __builtin_amdgcn_wmma_f32_16x16x4_f32(
            false, a, false, bb, (short)0, acc2[nt], false, false);
      }
    }
    __syncthreads();  // B2: all hs reads done, safe to overwrite as staging

    // stage results (+bias) into LDS for coalesced output
    #pragma unroll
    for (int nt = 0; nt < 4; ++nt) {
      #pragma unroll
      for (int i = 0; i < 8; ++i) {
        const int row = d0 + half * 8 + i;
        hs[row * HSTR + nt * 16 + n] = acc2[nt][i] + beff;
      }
    }
    __syncthreads();  // B3: staging ready

    // coalesced stores: each wave writes one full contiguous 256B row segment
    for (int i = tid; i < (C_ * TL) / 2; i += 256) {
      const int r  = i >> 5;
      const int c2 = (i & 31) << 1;
      const int l  = l0 + c2;
      const float vx = hs[r * HSTR + c2];
      const float vy = hs[r * HSTR + c2 + 1];
      const size_t idx = obase + (size_t)r * LP_ + l;
      if (l + 1 < LP_) {
        v2f v; v.x = vx; v.y = vy;
        *(v2f*)(out + idx) = v;            // 8B-aligned b64 store
      } else if (l < LP_) {
        out[idx] = vx;
      }
    }
  }
}

// ---------------------------------------------------------------------------
extern "C" void kernel_launch(void* const* d_in, const int* in_sizes, int n_in,
                              void* d_out, int out_size, void* d_ws, size_t ws_size,
                              hipStream_t stream) {
  const float* x    = (const float*)d_in[0];
  const float* nz   = (const float*)d_in[1];
  const float* wg   = (const float*)d_in[2];
  const float* wn   = (const float*)d_in[3];
  const float* w1   = (const float*)d_in[4];
  const float* b1   = (const float*)d_in[5];
  const float* w2   = (const float*)d_in[6];
  const float* b2   = (const float*)d_in[7];
  float* out = (float*)d_out;
  float* ws  = (float*)d_ws;

  const size_t smemBytes = (size_t)(C_ * W1STR + C_ + D_ * XSTR + C_ * HSTR) * sizeof(float);
  (void)hipFuncSetAttribute((const void*)fused_conv_kernel,
                            hipFuncAttributeMaxDynamicSharedMemorySize,
                            (int)smemBytes);

  gating_kernel<<<G_, 256, 0, stream>>>(x, nz, wg, wn, out, ws);
  loss_sum_kernel<<<1, 1, 0, stream>>>(ws, out);
  fused_conv_kernel<<<B_ * G_ * 8, 256, smemBytes, stream>>>(x, w1, b1, w2, b2, ws, out);
}